// FraudGraphQNNClassifier_65481071409525
// MI455X (gfx1250) — compile-verified
//
#include <hip/hip_runtime.h>
#include <math.h>

// ---------------------------------------------------------------------------
// FraudGraphQNNClassifier for MI455X (gfx1250)
//   Phase 1: serial scan of 8193 tiny layers (latency-bound, 1 lane, hw v_tanh)
//   Phase 2: fid = (S S^T)^2 via V_WMMA_F32_16X16X4_F32 (rank-2 GEMM, K padded)
//            one wave = 16x64 strip (4 tiles, shared A fragment)
//            below-diagonal tiles short-circuit to zero stores (triu mask)
//            268 MB output -> pure store-bandwidth bound (~11.5 us floor)
// ---------------------------------------------------------------------------

typedef __attribute__((ext_vector_type(2))) float v2f;
typedef __attribute__((ext_vector_type(8))) float v8f;

__device__ __forceinline__ float hw_tanh(float v) {
#if __has_builtin(__builtin_amdgcn_tanhf)
    return __builtin_amdgcn_tanhf(v);   // gfx1250 v_tanh_f32
#else
    return tanhf(v);
#endif
}

__device__ __forceinline__ float fid_weight(float d) {
    float f = d * d;                                    // squared overlap
    return (f >= 0.99f) ? 1.0f : ((f >= 0.8f) ? 0.5f : 0.0f);
}

// ---------------------------------------------------------------------------
// Phase 1: sequential chain. One lane owns h in registers; per-layer params
// arrive as 1x b128 + 3x b64 loads (328 KB stream, L2-resident) pipelined off
// the 4-deep dependent FMA/TANH chain. Emits unit-state table S[n] (float2)
// and the final logit.
// ---------------------------------------------------------------------------
__global__ void __launch_bounds__(32)
qnn_chain_kernel(const float* __restrict__ x,
                 const float* __restrict__ W,      // [L,2,2] row-major
                 const float* __restrict__ b,      // [L,2]
                 const float* __restrict__ sc,     // [L,2]
                 const float* __restrict__ sh,     // [L,2]
                 const float* __restrict__ Wout,   // [1,2]
                 const float* __restrict__ bout,   // [1]
                 int L,
                 float2* __restrict__ S,           // [L+1] unit states
                 float* __restrict__ out0)         // logits (1)
{
    if (threadIdx.x != 0) return;

    float h0 = x[0], h1 = x[1];

    float r = 1.0f / (sqrtf(h0 * h0 + h1 * h1) + 1e-12f);
    S[0] = make_float2(h0 * r, h1 * r);

    const float4* W4 = (const float4*)W;
    const float2* B2 = (const float2*)b;
    const float2* SC = (const float2*)sc;
    const float2* SH = (const float2*)sh;

    #pragma unroll 4
    for (int l = 0; l < L; ++l) {
        float4 w  = W4[l];
        float2 bb = B2[l];
        float2 ss = SC[l];
        float2 tt = SH[l];
        float u0 = fmaf(w.x, h0, fmaf(w.y, h1, bb.x));
        float u1 = fmaf(w.z, h0, fmaf(w.w, h1, bb.y));
        h0 = fmaf(hw_tanh(u0), ss.x, tt.x);
        h1 = fmaf(hw_tanh(u1), ss.y, tt.y);
        float inv = 1.0f / (sqrtf(h0 * h0 + h1 * h1) + 1e-12f);
        S[l + 1] = make_float2(h0 * inv, h1 * inv);
    }

    out0[0] = fmaf(Wout[0], h0, fmaf(Wout[1], h1, bout[0]));
}

// ---------------------------------------------------------------------------
// Phase 2: one wave per 16x64 strip (4 tiles sharing the A fragment).
// f32 WMMA fragment layout (CDNA5 ISA):
//   A 16x4 : lanes 0-15 hold M=lane {K0,K1}; lanes 16-31 hold {K2,K3} = pad 0
//   B 4x16 : lanes 0-15 hold N=lane {K0,K1}; lanes 16-31 hold {K2,K3} = pad 0
//   C/D    : VGPR v, lane L -> row = v + 8*(L>=16), col = L&15
// Tile classes (uniform per wave):
//   below diagonal (i0 >= j0+15)          -> store zeros, skip WMMA
//   above diagonal (i0+15 < j0), interior -> WMMA, no triangle mask
//   straddling / edge                     -> WMMA + mask (+ bounds on edge)
// ---------------------------------------------------------------------------
__global__ void __launch_bounds__(256)
qnn_fid_wmma_kernel(const float2* __restrict__ S,
                    float* __restrict__ E,   // n x n edge-weight matrix
                    int n, int ntiles, int ngroupsX)
{
    const int lane = threadIdx.x;                            // 0..31 (wave32)
    const int gX   = blockIdx.x;                             // strip of 4 tiles
    const int tY   = blockIdx.y * blockDim.y + threadIdx.y;  // tile row
    if (tY >= ntiles) return;                                // uniform per wave

    const int  i0    = tY << 4;
    const int  l16   = lane & 15;
    const bool hi    = lane >= 16;
    const int  rbase = i0 + (hi ? 8 : 0);

    // Row-base pointer: single 64-bit mul per wave, then p += n per row.
    float* const prow = E + (size_t)rbase * (size_t)n;

    // A fragment, shared by all 4 tiles in the strip.
    v2f a = {0.0f, 0.0f};
    if (!hi) {
        int ia = i0 + l16;
        if (ia < n) { float2 v = S[ia]; a.x = v.x; a.y = v.y; }
    }

    #pragma unroll
    for (int t = 0; t < 4; ++t) {
        const int tX = (gX << 2) + t;
        const int j0 = tX << 4;
        if (j0 >= n) break;                                  // uniform

        const int col  = j0 + l16;
        const bool edge = (i0 + 15 >= n) || (j0 + 15 >= n);
        float* p = prow + col;

        // --- fully below diagonal: triu mask zeroes everything ---
        if ((i0 >= j0 + 15) && !edge) {
            #pragma unroll
            for (int v = 0; v < 8; ++v) { *p = 0.0f; p += n; }
            continue;
        }

        // B fragment for this tile.
        v2f bf = {0.0f, 0.0f};
        if (!hi) {
            int jb = j0 + l16;
            if (jb < n) { float2 v = S[jb]; bf.x = v.x; bf.y = v.y; }
        }

        v8f c = {};
        c = __builtin_amdgcn_wmma_f32_16x16x4_f32(
                false, a, false, bf, (short)0, c, false, false);

        if (!edge) {
            if (i0 + 15 < j0) {
                // --- fully above diagonal: no mask, straight-line stores ---
                #pragma unroll
                for (int v = 0; v < 8; ++v) {
                    *p = fid_weight(c[v]);
                    p += n;
                }
            } else {
                // --- straddles diagonal: per-element triangle select ---
                #pragma unroll
                for (int v = 0; v < 8; ++v) {
                    float w = fid_weight(c[v]);
                    if (rbase + v >= col) w = 0.0f;
                    *p = w;
                    p += n;
                }
            }
        } else {
            // --- edge tiles: bounds-guarded (rare: last tile row/col) ---
            if (col < n) {
                #pragma unroll
                for (int v = 0; v < 8; ++v) {
                    int row = rbase + v;
                    if (row < n) {
                        float w = fid_weight(c[v]);
                        if (row >= col) w = 0.0f;
                        *p = w;
                    }
                    p += n;
                }
            }
        }
    }
}

// ---------------------------------------------------------------------------
// Launch. Inputs (setup_inputs order):
//   0:x[2] 1:W_layers[L,2,2] 2:b_layers[L,2] 3:scale[L,2] 4:shift[L,2]
//   5:W_out[1,2] 6:b_out[1]
// Output: [logits(1)] ++ [edge_weights(n*n)], n = L+1, float32.
// Workspace: n * sizeof(float2) (~64 KB) unit-state table.
// ---------------------------------------------------------------------------
extern "C" void kernel_launch(void* const* d_in, const int* in_sizes, int n_in,
                              void* d_out, int out_size, void* d_ws, size_t ws_size,
                              hipStream_t stream) {
    const float* x    = (const float*)d_in[0];
    const float* W    = (const float*)d_in[1];
    const float* b    = (const float*)d_in[2];
    const float* sc   = (const float*)d_in[3];
    const float* sh   = (const float*)d_in[4];
    const float* Wout = (const float*)d_in[5];
    const float* bout = (const float*)d_in[6];

    const int L = in_sizes[1] / 4;   // stacked layers (8193)
    const int n = L + 1;             // states (8194)

    float*  out_logits = (float*)d_out;
    float*  E          = (float*)d_out + 1;
    float2* S          = (float2*)d_ws;

    // Phase 1: serial scan (1 wave).
    qnn_chain_kernel<<<dim3(1), dim3(32), 0, stream>>>(
        x, W, b, sc, sh, Wout, bout, L, S, out_logits);

    // Phase 2: 8 waves per block, one 16x64 strip (4 tiles) per wave.
    const int ntiles   = (n + 15) >> 4;          // 513
    const int ngroupsX = (ntiles + 3) >> 2;      // 129
    dim3 block(32, 8);
    dim3 grid(ngroupsX, (ntiles + 7) / 8);
    qnn_fid_wmma_kernel<<<grid, block, 0, stream>>>(S, E, n, ntiles, ngroupsX);
}